// GridInterpolationLayer_4398046511666
// MI455X (gfx1250) — compile-verified
//
#include <hip/hip_runtime.h>

// Trilinear grid gather for MI455X (gfx1250, wave32).
// Pure bandwidth-bound gather/scatter: ~207 MB HBM traffic -> ~9 us floor.
// 64 threads/point (16 lanes x float4 per corner); B128 coalesced loads from
// the L2-resident 32 MB grid; non-temporal stores for the 174 MB streamed
// output; global_prefetch_b8 for the second corner set.
// v2: replaced exact f32 divisions by `cube` with multiply by 31.0f
// (1/fl(1/31) rounds to 31.0f) — removes 3 IEEE div sequences (~35 VALU ops).

typedef __attribute__((ext_vector_type(4))) float f4;

constexpr int BS = 4;      // batch
constexpr int NP = 20000;  // points per batch
constexpr int S  = 32;     // grid spatial size (d=h=w)
constexpr int C  = 64;     // code length

__global__ __launch_bounds__(256) void grid_interp_kernel(
    const float* __restrict__ grid,  // (BS, S, S, S, C)
    const float* __restrict__ pts,   // (BS, NP, 3)
    float* __restrict__ lat,         // (BS, NP, 8, C)
    float* __restrict__ wgt,         // (BS, NP, 8)
    float* __restrict__ xloc)        // (BS, NP, 8, 3)
{
  const int tid  = threadIdx.x;
  const int q    = tid >> 6;        // point slot within block (0..3)
  const int lane = tid & 63;        // 64 threads cooperate on one point
  const int pi   = blockIdx.x * 4 + q;  // global point index
  if (pi >= BS * NP) return;
  const int b = pi / NP;

  const float inv_cube = (float)(S - 1);  // 31.0f == 1/fl(1/31) to fp32

  // All 64 threads redundantly load the 3 coords (same cachelines, cheap).
  const float* pp = pts + (size_t)pi * 3;
  float pd = fminf(fmaxf(pp[0], 1e-6f), 1.0f - 1e-6f);
  float ph = fminf(fmaxf(pp[1], 1e-6f), 1.0f - 1e-6f);
  float pw = fminf(fmaxf(pp[2], 1e-6f), 1.0f - 1e-6f);

  // t = p/cube ; ind0 = floor(t) ; f = fractional position in cell
  float td = pd * inv_cube, th = ph * inv_cube, tw = pw * inv_cube;
  float i0d = floorf(td), i0h = floorf(th), i0w = floorf(tw);
  float fd = td - i0d, fh = th - i0h, fw = tw - i0w;
  int id = (int)i0d, ih = (int)i0h, iw = (int)i0w;

  // Corner mapping: k -> offsets ((k>>2)&1, (k>>1)&1, k&1) for (d,h,w)
  // 16 lanes per corner, each moving float4 (64 channels = 16 * 4).
  const int kc = lane >> 4;          // corner within half-set (0..3)
  const int c4 = (lane & 15) * 4;    // channel offset

  size_t gb[2];
#pragma unroll
  for (int s = 0; s < 2; ++s) {
    int k  = kc + 4 * s;
    int dd = id + ((k >> 2) & 1);
    int hh = ih + ((k >> 1) & 1);
    int ww = iw + (k & 1);
    gb[s] = ((((size_t)b * S + dd) * S + hh) * S + ww) * (size_t)C + c4;
  }
  // Prefetch second corner set (gfx1250 global_prefetch_b8).
  __builtin_prefetch(grid + gb[1], 0, 3);

  const size_t latBase = (size_t)pi * 8 * C;
#pragma unroll
  for (int s = 0; s < 2; ++s) {
    int k = kc + 4 * s;
    f4 v = *(const f4*)(grid + gb[s]);  // regular load: keep grid hot in L2
    // Non-temporal streaming store: lat is written once, never re-read.
    __builtin_nontemporal_store(v, (f4*)(lat + latBase + (size_t)k * C + c4));
  }

  // 8 lanes per point emit weight + xloc.
  if (lane < 8) {
    int k = lane;
    float od = (float)((k >> 2) & 1);
    float oh = (float)((k >> 1) & 1);
    float ow = (float)(k & 1);
    // weight_axis = |f - (1 - off)| ; weight = product over axes
    float wd  = fabsf(fd - (1.0f - od));
    float wh  = fabsf(fh - (1.0f - oh));
    float ww_ = fabsf(fw - (1.0f - ow));
    __builtin_nontemporal_store(wd * wh * ww_, wgt + (size_t)pi * 8 + k);
    // xloc = f - off per axis
    size_t xb = ((size_t)pi * 8 + k) * 3;
    __builtin_nontemporal_store(fd - od, xloc + xb + 0);
    __builtin_nontemporal_store(fh - oh, xloc + xb + 1);
    __builtin_nontemporal_store(fw - ow, xloc + xb + 2);
  }
}

extern "C" void kernel_launch(void* const* d_in, const int* in_sizes, int n_in,
                              void* d_out, int out_size, void* d_ws, size_t ws_size,
                              hipStream_t stream) {
  const float* grid = (const float*)d_in[0];  // 4*32*32*32*64 floats
  const float* pts  = (const float*)d_in[1];  // 4*20000*3 floats

  float* lat  = (float*)d_out;                       // (4,20000,8,64)
  float* wgt  = lat + (size_t)BS * NP * 8 * C;       // (4,20000,8)
  float* xlc  = wgt + (size_t)BS * NP * 8;           // (4,20000,8,3)

  const int npts   = BS * NP;          // 80000
  const int blocks = (npts + 3) / 4;   // 4 points per 256-thread block
  grid_interp_kernel<<<blocks, 256, 0, stream>>>(grid, pts, lat, wgt, xlc);
}